// Propagator_85444079387161
// MI455X (gfx1250) — compile-verified
//
#include <hip/hip_runtime.h>

// ---------------------------------------------------------------------------
// GGNN propagator for MI455X (gfx1250, wave32).
//   Phase 1: build a-buffer [B*N*L, 192] = [a_in | a_out | state_cur]
//   Phase 2: COO scatter with global_atomic_add_f32
//   Phase 3: LDS-staged fused r/z/h GEMMs via V_WMMA_F32_16X16X4_F32 + GRU
// ---------------------------------------------------------------------------

#define B_      8
#define NNODE   2000
#define NET     2
#define L_      8
#define D_      64
#define NNZ_    32000
#define EN_     (NET * NNODE)          // 4000
#define K3D     192                    // 3*D
#define ROWS    (B_ * NNODE * L_)      // 128000 rows of the big matrix
#define ABUF_ELEMS ((long)ROWS * K3D)  // 24,576,000 floats

typedef __attribute__((ext_vector_type(2))) float v2f;
typedef __attribute__((ext_vector_type(8))) float v8f;

// 16-byte payload type for the async b128 builtin (gcc-style vector, matches
// the builtin's declared parameter type exactly).
typedef int b128_t __attribute__((vector_size(16)));
typedef __attribute__((address_space(1))) b128_t* g128p;   // global ptr
typedef __attribute__((address_space(3))) b128_t* l128p;   // LDS ptr

// LDS A-tile row stride (floats): 196*4=784B -> rows stay 16B-aligned and
// rows are skewed by 4 banks, so 16 lanes reading one column pair across
// rows 0..15 hit 16 distinct banks.
#define LDS_STRIDE 196

#if defined(__has_builtin)
#if __has_builtin(__builtin_amdgcn_global_load_async_to_lds_b128)
#define HAVE_ASYNC_B128 1
#endif
#if __has_builtin(__builtin_amdgcn_s_wait_asynccnt)
#define HAVE_WAIT_ASYNC 1
#endif
#endif

// ---------------------------------------------------------------------------
// Kernel 1: a_buf[row][0:128] = 0 ; a_buf[row][128:192] = state_cur[row][:]
// ---------------------------------------------------------------------------
__global__ __launch_bounds__(256) void init_abuf(const float* __restrict__ state_cur,
                                                 float* __restrict__ abuf) {
  long i = (long)blockIdx.x * blockDim.x + threadIdx.x;   // grid sized exactly
  long row = i / K3D;
  int  c   = (int)(i - row * K3D);
  float v = 0.0f;
  if (c >= 2 * D_) v = state_cur[row * D_ + (c - 2 * D_)];
  abuf[i] = v;
}

// ---------------------------------------------------------------------------
// Kernel 2: COO scatter.  One wave per edge; each lane owns 16 of the 512
// floats of the L x D message row and issues 16 global_atomic_add_f32.
// ---------------------------------------------------------------------------
__global__ __launch_bounds__(256) void spmm_scatter(const float* __restrict__ state_in,
                                                    const float* __restrict__ state_out,
                                                    const float* __restrict__ vals,
                                                    const int*   __restrict__ rows,
                                                    const int*   __restrict__ cols,
                                                    float* __restrict__ abuf) {
  int wave = (int)((blockIdx.x * blockDim.x + threadIdx.x) >> 5);  // global wave id
  int lane = threadIdx.x & 31;

  int b   = wave / (NET * NNZ_);
  int rem = wave - b * (NET * NNZ_);
  int et  = rem / NNZ_;
  int e   = rem - et * NNZ_;
  int o   = (b * NET + et) * NNZ_ + e;

  float w = vals[o];
  int   r = rows[o];
  int   c = cols[o];

  const float* srcBase = (et == 0) ? state_in : state_out;
  const float* src = srcBase + ((long)b * EN_ + c) * (L_ * D_);        // 512 floats
  float* dstRow = abuf + ((long)(b * NNODE + r) * L_) * K3D + et * D_;

  int l     = lane >> 2;            // 0..7  (L index)
  int dbase = (lane & 3) * 16;      // 0,16,32,48 within D

  const float4* s4 = reinterpret_cast<const float4*>(src + l * D_ + dbase);
  float* dst = dstRow + l * K3D + dbase;

#pragma unroll
  for (int q = 0; q < 4; ++q) {
    float4 x = s4[q];
    atomicAdd(dst + q * 4 + 0, w * x.x);
    atomicAdd(dst + q * 4 + 1, w * x.y);
    atomicAdd(dst + q * 4 + 2, w * x.z);
    atomicAdd(dst + q * 4 + 3, w * x.w);
  }
}

// ---------------------------------------------------------------------------
// Kernel 3: fused GRU gates with FP32 WMMA, A-tiles staged in LDS.
// One wave owns a 16-row M-tile (= 2 nodes x L=8) of the [128000,192] matrix.
//
// A-fragment (ISA 7.12.2, 32-bit A 16x4): lanes 0-15 hold K={0,1}, lanes
// 16-31 hold K={2,3} -> float2 at LDS row (lane&15), col k0 + 2*(lane>>4).
// B-fragment (4x16): lane holds B[k0+2h+{0,1}][n0+lm] = W[n0+lm][k0+2h..],
// contiguous in W's row -> float2 global load (tiny, L2-resident).
// C/D tile: vgpr v, lanes 0-15 -> (M=v, N=lm); lanes 16-31 -> (M=v+8, N=lm).
//
// Flow: stage A (async-to-LDS if available); r&z GEMMs share A fragments in
// one K-loop; overwrite LDS cols 128..191 with r*state_cur (legal only after
// ALL r/z K-loops, which read full K, are done); h GEMM reads the joined
// matrix from LDS; final blend reads state_cur from global (L2 hit).
// ---------------------------------------------------------------------------
#define WPB 4   // waves per block (LDS: 4 * 16 * 196 * 4 = 50,176 B)

__global__ __launch_bounds__(128) void gru_fused(const float* __restrict__ abuf,
                                                 const float* __restrict__ Wr,
                                                 const float* __restrict__ br,
                                                 const float* __restrict__ Wz,
                                                 const float* __restrict__ bz,
                                                 const float* __restrict__ Wh,
                                                 const float* __restrict__ bh,
                                                 float* __restrict__ out) {
  __shared__ float ldsA[WPB][16][LDS_STRIDE];

  int wid  = threadIdx.x >> 5;
  int lane = threadIdx.x & 31;
  int hsel = lane >> 4;                  // 0: K pair {0,1}, 1: K pair {2,3}
  int lm   = lane & 15;

  int tile = blockIdx.x * WPB + wid;     // 0..7999
  long rowBase = (long)tile * 16;
  const float* A = abuf + rowBase * K3D;
  float* myA = &ldsA[wid][0][0];

  // Keep the tiny weight matrices hot (global_prefetch_b8).
  __builtin_prefetch(Wr + lm * K3D, 0, 1);
  __builtin_prefetch(Wz + lm * K3D, 0, 1);
  __builtin_prefetch(Wh + lm * K3D, 0, 1);

  // ---- stage the 16x192 A-tile into LDS (24 x 16B per lane) ----
#if HAVE_ASYNC_B128
#pragma unroll
  for (int i = 0; i < 24; ++i) {
    int f4  = lane + 32 * i;             // 0..767 float4 chunks (48 per row)
    int row = f4 / 48;
    int col = (f4 - row * 48) * 4;
    __builtin_amdgcn_global_load_async_to_lds_b128(
        (g128p)(A + row * K3D + col),
        (l128p)(myA + row * LDS_STRIDE + col),
        0, 0);
  }
#if HAVE_WAIT_ASYNC
  __builtin_amdgcn_s_wait_asynccnt(0);
#endif
#else
#pragma unroll
  for (int i = 0; i < 24; ++i) {
    int f4  = lane + 32 * i;
    int row = f4 / 48;
    int col = (f4 - row * 48) * 4;
    *(float4*)(myA + row * LDS_STRIDE + col) =
        *(const float4*)(A + row * K3D + col);
  }
#endif
  __syncthreads();

  const float* aLds = myA + lm * LDS_STRIDE + 2 * hsel;  // per-lane A-frag base

  // ---- r and z gates: shared A fragments, 2 WMMAs per K-slice ----
  v8f racc[4], zacc[4];
  for (int nt = 0; nt < 4; ++nt) {
    v8f ar = {};
    v8f az = {};
    const float* wBr = Wr + (nt * 16 + lm) * K3D + 2 * hsel;
    const float* wBz = Wz + (nt * 16 + lm) * K3D + 2 * hsel;
#pragma unroll 8
    for (int ks = 0; ks < 48; ++ks) {
      v2f af  = *(const v2f*)(aLds + ks * 4);
      v2f bfr = *(const v2f*)(wBr + ks * 4);
      v2f bfz = *(const v2f*)(wBz + ks * 4);
      ar = __builtin_amdgcn_wmma_f32_16x16x4_f32(false, af, false, bfr,
                                                 (short)0, ar, false, false);
      az = __builtin_amdgcn_wmma_f32_16x16x4_f32(false, af, false, bfz,
                                                 (short)0, az, false, false);
    }
    racc[nt] = ar;
    zacc[nt] = az;
  }

  // ---- activations; overwrite LDS cols 128..191 with r * state_cur ----
  for (int nt = 0; nt < 4; ++nt) {
    int   n     = nt * 16 + lm;
    float biasR = br[n];
    float biasZ = bz[n];
    v8f   ar = racc[nt];
    v8f   az = zacc[nt];
#pragma unroll
    for (int v = 0; v < 8; ++v) {
      int   m   = v + 8 * hsel;
      float rv  = 1.0f / (1.0f + __expf(-(ar[v] + biasR)));
      float cur = myA[m * LDS_STRIDE + 2 * D_ + n];
      myA[m * LDS_STRIDE + 2 * D_ + n] = rv * cur;   // joined[:,128:192]
      az[v] = 1.0f / (1.0f + __expf(-(az[v] + biasZ)));
    }
    zacc[nt] = az;
  }

  // ---- h gate: full K from LDS (now holds the joined matrix) ----
  for (int nt = 0; nt < 4; ++nt) {
    v8f acc = {};
    const float* wBh = Wh + (nt * 16 + lm) * K3D + 2 * hsel;
#pragma unroll 8
    for (int ks = 0; ks < 48; ++ks) {
      v2f af = *(const v2f*)(aLds + ks * 4);
      v2f bf = *(const v2f*)(wBh + ks * 4);
      acc = __builtin_amdgcn_wmma_f32_16x16x4_f32(false, af, false, bf,
                                                  (short)0, acc, false, false);
    }
    int   n    = nt * 16 + lm;
    float bias = bh[n];
    v8f   z    = zacc[nt];
#pragma unroll
    for (int v = 0; v < 8; ++v) {
      int   m    = v + 8 * hsel;
      float hhat = tanhf(acc[v] + bias);
      float cur  = A[m * K3D + 2 * D_ + n];          // L2-hot global read
      out[(rowBase + m) * D_ + n] = (1.0f - z[v]) * cur + z[v] * hhat;
    }
  }
}

// ---------------------------------------------------------------------------
extern "C" void kernel_launch(void* const* d_in, const int* in_sizes, int n_in,
                              void* d_out, int out_size, void* d_ws, size_t ws_size,
                              hipStream_t stream) {
  const float* state_in  = (const float*)d_in[0];
  const float* state_out = (const float*)d_in[1];
  const float* state_cur = (const float*)d_in[2];
  const float* A_vals    = (const float*)d_in[3];
  const int*   A_rows    = (const int*)d_in[4];
  const int*   A_cols    = (const int*)d_in[5];
  const float* W_r       = (const float*)d_in[6];
  const float* b_r       = (const float*)d_in[7];
  const float* W_z       = (const float*)d_in[8];
  const float* b_z       = (const float*)d_in[9];
  const float* W_h       = (const float*)d_in[10];
  const float* b_h       = (const float*)d_in[11];
  float* out  = (float*)d_out;
  float* abuf = (float*)d_ws;            // 24,576,000 floats (~98.3 MB)

  // Phase 1: build a-buffer (exact grid: 24,576,000 / 256 = 96,000 blocks)
  init_abuf<<<(unsigned)(ABUF_ELEMS / 256), 256, 0, stream>>>(state_cur, abuf);

  // Phase 2: scatter (B*2*NNZ = 512,000 waves -> 64,000 blocks of 8 waves)
  spmm_scatter<<<(B_ * NET * NNZ_) / 8, 256, 0, stream>>>(
      state_in, state_out, A_vals, A_rows, A_cols, abuf);

  // Phase 3: fused WMMA GRU (8000 M-tiles -> 2000 blocks of 4 waves)
  gru_fused<<<(ROWS / 16) / WPB, 128, 0, stream>>>(
      abuf, W_r, b_r, W_z, b_z, W_h, b_h, out);
}